// PolicyNet_78365973283198
// MI455X (gfx1250) — compile-verified
//
#include <hip/hip_runtime.h>
#include <cmath>
#include <stdint.h>

// ---------------------------------------------------------------------------
// PolicyNet forward for MI455X (gfx1250, wave32, WMMA + async-to-LDS DMA)
//   tempHS = tanh(state_HS @ W_fs + b_fs)        [50000 x 128]  <- f32 WMMA
//   u0     = sigmoid(tempHS @ W_fp + b_fp)       [50000]        <- fused epilogue
//   uk[e]  = sigmoid(<tempHS[seg[e]], hats[e]>)  [600000]       <- streaming gather
//   out    = concat(u0, uk)
// ---------------------------------------------------------------------------

typedef float v2f __attribute__((ext_vector_type(2)));
typedef float v4f __attribute__((ext_vector_type(4)));
typedef float v8f __attribute__((ext_vector_type(8)));

#define N_NODES 50000
#define N_EDGES 600000
#define IN_DIM  512
#define E_DIM   128

#define KC       32    // K-chunk staged through LDS
#define ASTRIDE  36    // padded row stride (floats): 144B rows, 16B aligned, conflict-free frags
#define WSTRIDE  136   // padded row stride for k-major W fallback path

// One 16-byte async DMA element per lane: copies 16 bytes at byte offset gofs_byte from base to LDS byte address lds_byte.
// gfx1250 GLOBAL_LOAD_ASYNC_TO_LDS_B128 (GVS form), tracked by ASYNCcnt.
__device__ __forceinline__ void async_copy16(uint32_t lds_byte, uint32_t gofs_byte,
                                             const float* __restrict__ base) {
    asm volatile("global_load_async_to_lds_b128 %0, %1, %2"
                 :: "v"(lds_byte), "v"(gofs_byte), "s"(base)
                 : "memory");
}
__device__ __forceinline__ void wait_async0() {
    asm volatile("s_wait_asynccnt 0" ::: "memory");
}

// ---------------------------------------------------------------------------
// One-time transpose: Wt[n][k] = W[k][n]  (128 x 512, coalesced writes)
// ---------------------------------------------------------------------------
__global__ __launch_bounds__(256) void transpose_w_kernel(
    const float* __restrict__ W, float* __restrict__ Wt)
{
    int idx = blockIdx.x * 256 + threadIdx.x;   // 0 .. 65535
    int n = idx >> 9;                           // 0..127
    int k = idx & 511;                          // 0..511
    Wt[idx] = W[k * E_DIM + n];
}

// ---------------------------------------------------------------------------
// Kernel 1: 128x128 output block per workgroup (8 waves x 16 rows), K chunked
// by 32 through double-buffered LDS filled by async DMA.
// TW=true : W pre-transposed (n-major) -> B frag is one contiguous ds_load_b64
// TW=false: W k-major (no scratch)     -> B frag is two strided b32 loads
// f32 WMMA 16x16x4 layouts (ISA 7.12.2):
//   A  (16x4):  lane m=(l&15); vgpr0 = A[m][k+2*hi], vgpr1 = A[m][k+1+2*hi]
//   B  (4x16):  lane n=(l&15); vgpr0 = B[k+2*hi][n], vgpr1 = B[k+1+2*hi][n]
//   C/D(16x16): vgpr r: lanes 0-15 -> M=r, lanes 16-31 -> M=r+8; N=(l&15)
// ---------------------------------------------------------------------------
template <bool TW>
__global__ __launch_bounds__(256) void fs_fp_wmma_kernel(
    const float* __restrict__ A,      // state_HS [N_NODES][IN_DIM]
    const float* __restrict__ W,      // TW ? Wt[E_DIM][IN_DIM] : W_fs[IN_DIM][E_DIM]
    const float* __restrict__ bfs,    // b_fs [E_DIM]
    const float* __restrict__ wfp,    // W_fp [E_DIM]
    const float* __restrict__ bfp,    // b_fp [1]
    float*       __restrict__ tempHS, // [N_NODES][E_DIM]
    float*       __restrict__ u0_out) // [N_NODES]
{
    constexpr int WSZ = TW ? (E_DIM * ASTRIDE) : (KC * WSTRIDE);
    __shared__ float Alds[2][128 * ASTRIDE];  // 2 x 18.0 KB
    __shared__ float Wlds[2][WSZ];            // 2 x 18.0 KB (TW) / 2 x 17.0 KB

    const int tid   = threadIdx.x;
    const int lane  = tid & 31;
    const int wave  = tid >> 5;
    const int hi    = lane >> 4;
    const int lm    = lane & 15;
    const int mBase = blockIdx.x * 128;

    // ---- async DMA stage of one K-chunk into LDS buffer `buf` ----
    auto stage = [&](int buf, int kc) {
        const uint32_t wb = (uint32_t)(uintptr_t)&Wlds[buf][0];
        const uint32_t ab = (uint32_t)(uintptr_t)&Alds[buf][0];
        #pragma unroll
        for (int i = 0; i < 4; ++i) {
            int s = tid + i * 256;              // 0..1023 (1024 x 16B segments)
            if constexpr (TW) {
                int n = s >> 3, j = s & 7;      // Wt rows: 128 x KC floats
                async_copy16(wb + (uint32_t)((n * ASTRIDE + j * 4) * 4),
                             (uint32_t)(((uint32_t)n * IN_DIM + kc + j * 4) * 4), W);
            } else {
                int kr = s >> 5, j = s & 31;    // W rows: KC x 128 floats
                async_copy16(wb + (uint32_t)((kr * WSTRIDE + j * 4) * 4),
                             (uint32_t)(((kc + kr) * E_DIM + j * 4) * 4), W);
            }
        }
        #pragma unroll
        for (int i = 0; i < 4; ++i) {
            int s   = tid + i * 256;
            int row = s >> 3, j = s & 7;        // A rows: 128 x KC floats
            int grw = mBase + row; if (grw >= N_NODES) grw = N_NODES - 1;
            async_copy16(ab + (uint32_t)((row * ASTRIDE + j * 4) * 4),
                         (uint32_t)(((uint32_t)grw * IN_DIM + kc + j * 4) * 4), A);
        }
    };

    v8f acc[8];
    #pragma unroll
    for (int nt = 0; nt < 8; ++nt)
        #pragma unroll
        for (int i = 0; i < 8; ++i) acc[nt][i] = 0.0f;

    // prologue: DMA chunk 0
    stage(0, 0);
    wait_async0();
    __syncthreads();

    int cur = 0;
    for (int kc = 0; kc < IN_DIM; kc += KC) {
        // overlap: DMA next chunk into the other buffer while we run the matrix pipe
        if (kc + KC < IN_DIM) stage(cur ^ 1, kc + KC);

        const float* arow = &Alds[cur][(wave * 16 + lm) * ASTRIDE];
        const float* wch  = &Wlds[cur][0];
        #pragma unroll
        for (int ks = 0; ks < KC; ks += 4) {
            const int ka = ks + 2 * hi;
            v2f afrag = *(const v2f*)(arow + ka);
            v2f bfr[8];
            #pragma unroll
            for (int nt = 0; nt < 8; ++nt) {
                if constexpr (TW) {
                    // contiguous in k: single aligned 8B LDS load, no packing
                    bfr[nt] = *(const v2f*)(wch + (nt * 16 + lm) * ASTRIDE + ka);
                } else {
                    bfr[nt].x = wch[ ka      * WSTRIDE + nt * 16 + lm];
                    bfr[nt].y = wch[(ka + 1) * WSTRIDE + nt * 16 + lm];
                }
            }
            #pragma unroll
            for (int nt = 0; nt < 8; ++nt) {
                acc[nt] = __builtin_amdgcn_wmma_f32_16x16x4_f32(
                    false, afrag, false, bfr[nt], (short)0, acc[nt], false, false);
            }
        }
        wait_async0();      // next-chunk DMA complete
        __syncthreads();    // everyone done reading `cur`, next buffer published
        cur ^= 1;
    }

    // ---- epilogue: bias + tanh, store tempHS, fused GEMV (W_fp) + sigmoid ----
    float bfr[8], wfr[8];
    #pragma unroll
    for (int nt = 0; nt < 8; ++nt) {
        bfr[nt] = bfs[nt * 16 + lm];
        wfr[nt] = wfp[nt * 16 + lm];
    }
    const float bfp0 = bfp[0];

    #pragma unroll
    for (int r = 0; r < 8; ++r) {
        const int grow = mBase + wave * 16 + r + 8 * hi;  // C layout: M = r + 8*hi
        float psum = 0.0f;
        #pragma unroll
        for (int nt = 0; nt < 8; ++nt) {
            float h = tanhf(acc[nt][r] + bfr[nt]);
            psum += h * wfr[nt];
            if (grow < N_NODES)
                tempHS[(size_t)grow * E_DIM + nt * 16 + lm] = h;
        }
        #pragma unroll
        for (int m = 1; m < 16; m <<= 1) psum += __shfl_xor(psum, m, 32);
        if (lm == 0 && grow < N_NODES)
            u0_out[grow] = 1.0f / (1.0f + __expf(-(psum + bfp0)));
    }
}

// ---------------------------------------------------------------------------
// Kernel 2: one wave per edge. 128-dim dot = one float4 per lane (coalesced
// 512B per wave from hats; tempHS gathers served from L2: 25.6 MB << 192 MB,
// seg sorted -> locality). 32-lane butterfly reduce, lane 0 writes sigmoid.
// ---------------------------------------------------------------------------
__global__ __launch_bounds__(256) void edge_dot_kernel(
    const float* __restrict__ tempHS,
    const float* __restrict__ hats,
    const int*   __restrict__ seg,
    float*       __restrict__ uk_out)
{
    const int lane = threadIdx.x & 31;
    const int wave = threadIdx.x >> 5;
    const long long e = (long long)blockIdx.x * 8 + wave;
    if (e >= N_EDGES) return;

    const int node = seg[e];
    v4f hv = *(const v4f*)(tempHS + (size_t)node * E_DIM + lane * 4);
    v4f gv = *(const v4f*)(hats   + (size_t)e    * E_DIM + lane * 4);
    float d = hv.x * gv.x + hv.y * gv.y + hv.z * gv.z + hv.w * gv.w;
    #pragma unroll
    for (int m = 1; m < 32; m <<= 1) d += __shfl_xor(d, m, 32);
    if (lane == 0)
        uk_out[e] = 1.0f / (1.0f + __expf(-d));
}

// ---------------------------------------------------------------------------
extern "C" void kernel_launch(void* const* d_in, const int* in_sizes, int n_in,
                              void* d_out, int out_size, void* d_ws, size_t ws_size,
                              hipStream_t stream) {
    const float* state_HS = (const float*)d_in[0];
    const float* hats     = (const float*)d_in[1];
    const int*   seg      = (const int*)  d_in[2];
    const float* W_fs     = (const float*)d_in[3];
    const float* b_fs     = (const float*)d_in[4];
    const float* W_fp     = (const float*)d_in[5];
    const float* b_fp     = (const float*)d_in[6];
    float*       out      = (float*)d_out;

    float* tempHS = (float*)d_ws;                       // 25.6 MB
    const size_t tempHS_elems = (size_t)N_NODES * E_DIM;
    const size_t need_wt = (tempHS_elems + (size_t)IN_DIM * E_DIM) * sizeof(float);

    const int gridA = (N_NODES + 127) / 128;            // 391 workgroups
    if (ws_size >= need_wt) {
        float* Wt = tempHS + tempHS_elems;              // +256 KB
        transpose_w_kernel<<<256, 256, 0, stream>>>(W_fs, Wt);
        fs_fp_wmma_kernel<true><<<gridA, 256, 0, stream>>>(
            state_HS, Wt, b_fs, W_fp, b_fp, tempHS, out);
    } else {
        fs_fp_wmma_kernel<false><<<gridA, 256, 0, stream>>>(
            state_HS, W_fs, b_fs, W_fp, b_fp, tempHS, out);
    }

    const int gridB = (N_EDGES + 7) / 8;                // 75000 workgroups
    edge_dot_kernel<<<gridB, 256, 0, stream>>>(
        tempHS, hats, seg, out + N_NODES);
}